// RegionLoss_43430709297277
// MI455X (gfx1250) — compile-verified
//
#include <hip/hip_runtime.h>
#include <stdint.h>

// Problem constants (from reference)
#define NB 32
#define NA 5
#define NC 20
#define NH 76
#define NW 76
#define NCH   25                  // 5 + NC input channels per anchor
#define NCOL  29                  // 4 (pred box) + 5 (fields) + NC output cols
#define TILE_H 4                  // 76 % 4 == 0 -> 19 tiles per (b,a)
#define NTILES (NH / TILE_H)      // 19
#define TILE_ELEMS (TILE_H * NW)  // 304 spatial cells per block
#define LDS_FLOATS (TILE_ELEMS * NCOL)  // 8816 floats = 35,264 bytes
#define STORE_VEC4 (LDS_FLOATS / 4)     // 2204 b128 stores per block
#define THREADS 256
#define CH_STRIDE_B ((unsigned)(NH * NW * 4))   // 23,104 bytes between channels

__device__ __forceinline__ float sigmoidf_fast(float v) {
    return 1.0f / (1.0f + __expf(-v));
}

__global__ __launch_bounds__(THREADS)
void region_fuse_kernel(const float* __restrict__ in,
                        const float* __restrict__ anchors,
                        float* __restrict__ out)
{
    __shared__ float tile[LDS_FLOATS];

    const int tid = threadIdx.x;
    const int blk = blockIdx.x;
    const int t   = blk % NTILES;
    const int a   = (blk / NTILES) % NA;
    const int b   = blk / (NTILES * NA);
    const int h0  = t * TILE_H;

    // Base of this block's input slab: in[b][a*NCH + 0][h0][0]
    const float* gbase = in + ((size_t)b * (NA * NCH) + (size_t)a * NCH) * (size_t)(NH * NW)
                            + (size_t)h0 * NW;

    // LDS byte offset of tile[] (low 32 bits of the flat address == LDS offset)
    const unsigned lds_base = (unsigned)(uintptr_t)(&tile[0]);

    // ---------------- Phase 1: async DMA global -> LDS, transposing ----------------
    // Cell p = r*NW + w; channel c lands in column c+4 of LDS row p.
    // Outer loop over thread-owned cells, inner unrolled loop over channels with
    // purely additive addressing (no div/mod in the hot path). Lanes within a wave
    // hold consecutive p -> consecutive global addresses per channel (coalesced).
    for (int p = tid; p < TILE_ELEMS; p += THREADS) {
        unsigned voff  = (unsigned)(p * 4);                            // global byte offset
        unsigned laddr = lds_base + (unsigned)((p * NCOL + 4) * 4);    // LDS byte address
#pragma unroll
        for (int c = 0; c < NCH; ++c) {
            asm volatile("global_load_async_to_lds_b32 %0, %1, %2 offset:0 th:TH_LOAD_NT"
                         :: "v"(laddr), "v"(voff), "s"(gbase) : "memory");
            voff  += CH_STRIDE_B;
            laddr += 4u;
        }
    }
    asm volatile("s_wait_asynccnt 0" ::: "memory");
    __syncthreads();

    // ---------------- Phase 2: in-LDS decode of box / fields ----------------
    const float aw = anchors[2 * a + 0];
    const float ah = anchors[2 * a + 1];
    for (int p = tid; p < TILE_ELEMS; p += THREADS) {
        float* row = &tile[p * NCOL];
        const int w_idx = p % NW;
        const int h_idx = h0 + p / NW;

        const float r0 = row[4];   // raw x
        const float r1 = row[5];   // raw y
        const float r2 = row[6];   // raw w (stays raw in col 6)
        const float r3 = row[7];   // raw h (stays raw in col 7)
        const float r4 = row[8];   // raw conf

        const float sx = sigmoidf_fast(r0);
        const float sy = sigmoidf_fast(r1);
        const float sc = sigmoidf_fast(r4);

        row[0] = sx + (float)w_idx;        // bx
        row[1] = sy + (float)h_idx;        // by
        row[2] = __expf(r2) * aw;          // bw
        row[3] = __expf(r3) * ah;          // bh
        row[4] = sx;
        row[5] = sy;
        row[8] = sc;
    }
    __syncthreads();

    // ---------------- Phase 3: async DMA LDS -> global, fully coalesced b128 ----------------
    // Output region for this block is contiguous and 16B aligned:
    // rows [((b*NA+a)*NH+h0)*NW , +TILE_ELEMS) x NCOL floats = 35,264 bytes.
    float* obase = out + (size_t)(((b * NA + a) * NH + h0) * NW) * NCOL;
    for (int i = tid; i < STORE_VEC4; i += THREADS) {
        const unsigned goff  = (unsigned)(i * 16);
        const unsigned laddr = lds_base + (unsigned)(i * 16);
        asm volatile("global_store_async_from_lds_b128 %0, %1, %2 offset:0"
                     :: "v"(goff), "v"(laddr), "s"(obase) : "memory");
    }
    // S_ENDPGM performs an implicit wait-idle, covering outstanding ASYNCcnt.
}

extern "C" void kernel_launch(void* const* d_in, const int* in_sizes, int n_in,
                              void* d_out, int out_size, void* d_ws, size_t ws_size,
                              hipStream_t stream) {
    (void)in_sizes; (void)n_in; (void)d_ws; (void)ws_size; (void)out_size;
    const float* output  = (const float*)d_in[0];   // (32,125,76,76) f32
    const float* anchors = (const float*)d_in[1];   // (5,2) f32
    // d_in[2] (target) is unused by the reference's return value.
    float* out = (float*)d_out;                     // (924160, 29) f32

    const int grid = NB * NA * NTILES;              // 3040 blocks
    region_fuse_kernel<<<grid, THREADS, 0, stream>>>(output, anchors, out);
}